// GSModel_69612829934110
// MI455X (gfx1250) — compile-verified
//
#include <hip/hip_runtime.h>
#include <math.h>

#define NG      2048
#define IMG_W   160
#define IMG_H   160
#define NPIX    (IMG_W * IMG_H)

#define SH_C0f      0.28209479177387814f
#define SH_C1f      0.4886025119029199f
#define EPS2Df      0.3f
#define NEAR_PLANEf 0.01f
#define FAR_PLANEf  1.0e10f
#define ALPHA_MINf  (1.0f / 255.0f)
#define ALPHA_MAXf  0.999f
#define T_EPSf      1.0e-4f

typedef __attribute__((ext_vector_type(2))) float v2f;
typedef __attribute__((ext_vector_type(8))) float v8f;

// ---------------------------------------------------------------------------
// Phase 1: per-gaussian preprocessing.
// raw[i*8]    = { mx, my, cA, cB, cC, opac_masked, tz, 0 }
// rawcol[i*4] = { r, g, b, 0 }
// ---------------------------------------------------------------------------
__global__ __launch_bounds__(256) void gs_preprocess(
    const float* __restrict__ means3d, const float* __restrict__ opacities,
    const float* __restrict__ scales,  const float* __restrict__ quats,
    const float* __restrict__ sh0,     const float* __restrict__ shN,
    const float* __restrict__ c2w,     const float* __restrict__ Ks,
    float* __restrict__ raw, float* __restrict__ rawcol)
{
    int i = blockIdx.x * blockDim.x + threadIdx.x;
    if (i >= NG) return;

    // camera: viewmat = inv(camtoworld) for a rigid pose: [Rc^T | -Rc^T t]
    float R00 = c2w[0], R01 = c2w[1], R02 = c2w[2],  ctx = c2w[3];
    float R10 = c2w[4], R11 = c2w[5], R12 = c2w[6],  cty = c2w[7];
    float R20 = c2w[8], R21 = c2w[9], R22 = c2w[10], ctz = c2w[11];
    float W00 = R00, W01 = R10, W02 = R20;
    float W10 = R01, W11 = R11, W12 = R21;
    float W20 = R02, W21 = R12, W22 = R22;
    float t0 = -(W00 * ctx + W01 * cty + W02 * ctz);
    float t1 = -(W10 * ctx + W11 * cty + W12 * ctz);
    float t2 = -(W20 * ctx + W21 * cty + W22 * ctz);
    float fx = Ks[0], cx = Ks[2], fy = Ks[4], cy = Ks[5];

    float mx3 = means3d[i * 3 + 0];
    float my3 = means3d[i * 3 + 1];
    float mz3 = means3d[i * 3 + 2];

    // SH -> color
    float ddx = mx3 - ctx, ddy = my3 - cty, ddz = mz3 - ctz;
    float nrm = sqrtf(ddx * ddx + ddy * ddy + ddz * ddz);
    float inv = 1.0f / nrm;
    float dxn = ddx * inv, dyn = ddy * inv, dzn = ddz * inv;
    float col[3];
#pragma unroll
    for (int c = 0; c < 3; ++c) {
        float s0 = sh0[i * 3 + c];
        float s1 = shN[i * 9 + 0 + c];
        float s2 = shN[i * 9 + 3 + c];
        float s3 = shN[i * 9 + 6 + c];
        float v = SH_C0f * s0 + SH_C1f * (-dyn * s1 + dzn * s2 - dxn * s3) + 0.5f;
        col[c] = fmaxf(v, 0.0f);
    }

    // quat -> R, M = R * diag(s), cov3d = M M^T
    float qw = quats[i * 4 + 0], qx = quats[i * 4 + 1];
    float qy = quats[i * 4 + 2], qz = quats[i * 4 + 3];
    float qinv = 1.0f / sqrtf(qw * qw + qx * qx + qy * qy + qz * qz);
    qw *= qinv; qx *= qinv; qy *= qinv; qz *= qinv;
    float r00 = 1.0f - 2.0f * (qy * qy + qz * qz);
    float r01 = 2.0f * (qx * qy - qw * qz);
    float r02 = 2.0f * (qx * qz + qw * qy);
    float r10 = 2.0f * (qx * qy + qw * qz);
    float r11 = 1.0f - 2.0f * (qx * qx + qz * qz);
    float r12 = 2.0f * (qy * qz - qw * qx);
    float r20 = 2.0f * (qx * qz - qw * qy);
    float r21 = 2.0f * (qy * qz + qw * qx);
    float r22 = 1.0f - 2.0f * (qx * qx + qy * qy);
    float sx = scales[i * 3 + 0], sy = scales[i * 3 + 1], sz = scales[i * 3 + 2];
    float m00 = r00 * sx, m01 = r01 * sy, m02 = r02 * sz;
    float m10 = r10 * sx, m11 = r11 * sy, m12 = r12 * sz;
    float m20 = r20 * sx, m21 = r21 * sy, m22 = r22 * sz;
    float C00 = m00 * m00 + m01 * m01 + m02 * m02;
    float C01 = m00 * m10 + m01 * m11 + m02 * m12;
    float C02 = m00 * m20 + m01 * m21 + m02 * m22;
    float C11 = m10 * m10 + m11 * m11 + m12 * m12;
    float C12 = m10 * m20 + m11 * m21 + m12 * m22;
    float C22 = m20 * m20 + m21 * m21 + m22 * m22;

    // camera-space mean
    float tcx = W00 * mx3 + W01 * my3 + W02 * mz3 + t0;
    float tcy = W10 * mx3 + W11 * my3 + W12 * mz3 + t1;
    float tcz = W20 * mx3 + W21 * my3 + W22 * mz3 + t2;
    float rz = 1.0f / tcz;

    float limx = 1.3f * (0.5f * (float)IMG_W / fx);
    float limy = 1.3f * (0.5f * (float)IMG_H / fy);
    float txz = fminf(limx, fmaxf(-limx, tcx * rz));
    float tyz = fminf(limy, fmaxf(-limy, tcy * rz));

    // V = J * W3 (2x3)
    float frz = fx * rz, grz = fy * rz;
    float v00 = frz * (W00 - txz * W20);
    float v01 = frz * (W01 - txz * W21);
    float v02 = frz * (W02 - txz * W22);
    float v10 = grz * (W10 - tyz * W20);
    float v11 = grz * (W11 - tyz * W21);
    float v12 = grz * (W12 - tyz * W22);

    // cov2d = V * cov3d * V^T
    float u0 = C00 * v00 + C01 * v01 + C02 * v02;
    float u1 = C01 * v00 + C11 * v01 + C12 * v02;
    float u2 = C02 * v00 + C12 * v01 + C22 * v02;
    float p0 = C00 * v10 + C01 * v11 + C02 * v12;
    float p1 = C01 * v10 + C11 * v11 + C12 * v12;
    float p2 = C02 * v10 + C12 * v11 + C22 * v12;
    float a  = v00 * u0 + v01 * u1 + v02 * u2 + EPS2Df;
    float bb = v10 * u0 + v11 * u1 + v12 * u2;
    float cc = v10 * p0 + v11 * p1 + v12 * p2 + EPS2Df;

    float det  = a * cc - bb * bb;
    float dets = (det > 0.0f) ? det : 1.0f;
    float cA = cc / dets, cB = -bb / dets, cC = a / dets;

    float mxp = fx * tcx * rz + cx;
    float myp = fy * tcy * rz + cy;

    bool valid = (tcz > NEAR_PLANEf) && (tcz < FAR_PLANEf) && (det > 0.0f);
    float opac = 1.0f / (1.0f + expf(-opacities[i]));
    float opacM = valid ? opac : 0.0f;

    raw[i * 8 + 0] = mxp;  raw[i * 8 + 1] = myp;
    raw[i * 8 + 2] = cA;   raw[i * 8 + 3] = cB;
    raw[i * 8 + 4] = cC;   raw[i * 8 + 5] = opacM;
    raw[i * 8 + 6] = tcz;  raw[i * 8 + 7] = 0.0f;
    rawcol[i * 4 + 0] = col[0];
    rawcol[i * 4 + 1] = col[1];
    rawcol[i * 4 + 2] = col[2];
    rawcol[i * 4 + 3] = 0.0f;
}

// ---------------------------------------------------------------------------
// Phase 2: bitonic argsort of tz (ascending), single workgroup, LDS.
// ---------------------------------------------------------------------------
__global__ __launch_bounds__(1024) void gs_sort(const float* __restrict__ raw,
                                                int* __restrict__ order)
{
    __shared__ float key[NG];
    __shared__ int   idx[NG];
    int tid = threadIdx.x;
    for (int i = tid; i < NG; i += blockDim.x) {
        key[i] = raw[i * 8 + 6];
        idx[i] = i;
    }
    __syncthreads();
    for (int k = 2; k <= NG; k <<= 1) {
        for (int j = k >> 1; j > 0; j >>= 1) {
            for (int i = tid; i < NG; i += blockDim.x) {
                int ixj = i ^ j;
                if (ixj > i) {
                    bool up = ((i & k) == 0);
                    float ki = key[i], kj = key[ixj];
                    bool sw = up ? (ki > kj) : (ki < kj);
                    if (sw) {
                        key[i] = kj; key[ixj] = ki;
                        int t = idx[i]; idx[i] = idx[ixj]; idx[ixj] = t;
                    }
                }
            }
            __syncthreads();
        }
    }
    for (int i = tid; i < NG; i += blockDim.x) order[i] = idx[i];
}

// ---------------------------------------------------------------------------
// Phase 3: gather into sorted order; build WMMA B-matrix rows
// sp[s*8]   = { mx, my, cA, cB, cC, opac, tz, 0 }   (sorted)
// Bmat[s*16]= { r, g, b, tz, 1, 0, ..., 0 }          (channel-major row)
// ---------------------------------------------------------------------------
__global__ __launch_bounds__(256) void gs_reorder(
    const float* __restrict__ raw, const float* __restrict__ rawcol,
    const int* __restrict__ order,
    float* __restrict__ sp, float* __restrict__ Bmat)
{
    int s = blockIdx.x * blockDim.x + threadIdx.x;
    if (s >= NG) return;
    int g = order[s];
    const float4* rv = (const float4*)raw;
    float4 p0 = rv[g * 2 + 0];
    float4 p1 = rv[g * 2 + 1];
    float4* sv = (float4*)sp;
    sv[s * 2 + 0] = p0;
    sv[s * 2 + 1] = p1;
    float r = rawcol[g * 4 + 0];
    float gg = rawcol[g * 4 + 1];
    float b = rawcol[g * 4 + 2];
    float* B = Bmat + s * 16;
    B[0] = r; B[1] = gg; B[2] = b; B[3] = p1.z /*tz*/; B[4] = 1.0f;
#pragma unroll
    for (int c = 5; c < 16; ++c) B[c] = 0.0f;
}

// ---------------------------------------------------------------------------
// Phase 4: tiled alpha-compositing with WMMA f32 16x16x4 accumulation.
// One wave32 renders 16 pixels. Lane l: pixel m = l%16 (A-matrix row),
// B-matrix column n = l%16. Lane half h = l/16 owns gaussians {2h, 2h+1}
// of each 4-gaussian chunk (A-matrix K layout for f32 16x16x4).
// D[m][n] accumulates sum_k w[m,k] * {r,g,b,tz,1}[n][k] fully in VGPRs.
// ---------------------------------------------------------------------------
__device__ __forceinline__ void gs_chunk(
    const float4* __restrict__ spv, const float* __restrict__ Bmat,
    int c0, int h, int m, float pxc, float pyc, float& T, v8f& acc)
{
    int g0 = c0 + (h << 1);
    float4 P0 = spv[g0 * 2 + 0];         // mx my cA cB   (gaussian g0)
    float4 P1 = spv[g0 * 2 + 1];         // cC opac tz -
    float4 Q0 = spv[(g0 + 1) * 2 + 0];   // gaussian g0+1
    float4 Q1 = spv[(g0 + 1) * 2 + 1];
    float b0 = Bmat[(g0 << 4) + m];      // B row K=g0, col n=m
    float b1 = Bmat[((g0 + 1) << 4) + m];

    float dx = pxc - P0.x, dy = pyc - P0.y;
    float sig = 0.5f * (P0.z * dx * dx + P1.x * dy * dy) + P0.w * dx * dy;
    float al  = fminf(ALPHA_MAXf, P1.y * __expf(-sig));
    float a0  = (sig >= 0.0f && al >= ALPHA_MINf) ? al : 0.0f;

    dx = pxc - Q0.x; dy = pyc - Q0.y;
    sig = 0.5f * (Q0.z * dx * dx + Q1.x * dy * dy) + Q0.w * dx * dy;
    al  = fminf(ALPHA_MAXf, Q1.y * __expf(-sig));
    float a1 = (sig >= 0.0f && al >= ALPHA_MINf) ? al : 0.0f;

    float om0 = 1.0f - a0, om1 = 1.0f - a1;
    float pself  = om0 * om1;
    float pother = __shfl_xor(pself, 16, 32);

    // exclusive transmittance for this lane's two gaussians
    float Tk0 = h ? (T * pother) : T;
    float Tk1 = Tk0 * om0;
    float w0 = (Tk0 > T_EPSf) ? (a0 * Tk0) : 0.0f;
    float w1 = (Tk1 > T_EPSf) ? (a1 * Tk1) : 0.0f;

    // identical update on both half-lanes keeps per-pixel T bit-equal
    float plo = h ? pother : pself;
    float phi = h ? pself : pother;
    T = T * (plo * phi);

    v2f Aw = {w0, w1};
    v2f Bc = {b0, b1};
    acc = __builtin_amdgcn_wmma_f32_16x16x4_f32(
        false, Aw, false, Bc, (short)0, acc, false, false);
}

__global__ __launch_bounds__(256) void gs_render(
    const float* __restrict__ sp, const float* __restrict__ Bmat,
    float* __restrict__ out)
{
    int lane = threadIdx.x & 31;
    int wave = blockIdx.x * (blockDim.x >> 5) + (threadIdx.x >> 5);
    int m = lane & 15;
    int h = lane >> 4;

    int pix0 = wave * 16;               // 1600 waves * 16 px = 25600
    int pix  = pix0 + m;
    float pxc = (float)(pix % IMG_W) + 0.5f;
    float pyc = (float)(pix / IMG_W) + 0.5f;

    const float4* spv = (const float4*)sp;
    v8f acc = {0.f, 0.f, 0.f, 0.f, 0.f, 0.f, 0.f, 0.f};
    float T = 1.0f;

    // 16 gaussians (4 WMMA chunks) per outer iteration; amortized early-exit.
    for (int c0 = 0; c0 < NG; c0 += 16) {
#pragma unroll
        for (int u = 0; u < 4; ++u)
            gs_chunk(spv, Bmat, c0 + 4 * u, h, m, pxc, pyc, T, acc);
        if (!__any(T > T_EPSf)) break;   // wave-uniform early exit
    }

    // D layout: VGPR r -> pixel row M = r + 8*h, column N = lane%16
    int n = m;
#pragma unroll
    for (int r = 0; r < 8; ++r) {
        float val = acc[r];
        float av  = __shfl(acc[r], (lane & 16) + 4, 32);  // alpha_tot channel
        int p = pix0 + r + 8 * h;
        if (n < 3) {
            out[p * 4 + n] = val;                          // rgb
        } else if (n == 3) {
            out[p * 4 + 3] = val / fmaxf(av, 1e-10f);      // expected depth
        } else if (n == 4) {
            out[NPIX * 4 + p] = val;                       // alpha
        }
    }
}

// ---------------------------------------------------------------------------
extern "C" void kernel_launch(void* const* d_in, const int* in_sizes, int n_in,
                              void* d_out, int out_size, void* d_ws, size_t ws_size,
                              hipStream_t stream)
{
    (void)in_sizes; (void)n_in; (void)out_size; (void)ws_size;
    const float* means3d   = (const float*)d_in[0];
    const float* opacities = (const float*)d_in[1];
    const float* scales    = (const float*)d_in[2];
    const float* quats     = (const float*)d_in[3];
    const float* sh0       = (const float*)d_in[4];
    const float* shN       = (const float*)d_in[5];
    const float* c2w       = (const float*)d_in[6];
    const float* Ks        = (const float*)d_in[7];
    float* out = (float*)d_out;

    float* ws     = (float*)d_ws;
    float* raw    = ws;                   // NG*8  floats
    float* rawcol = ws + NG * 8;          // NG*4
    int*   order  = (int*)(ws + NG * 12); // NG ints
    float* sp     = ws + NG * 13;         // NG*8
    float* Bmat   = ws + NG * 21;         // NG*16  (total NG*37 floats ~ 303 KB)

    gs_preprocess<<<(NG + 255) / 256, 256, 0, stream>>>(
        means3d, opacities, scales, quats, sh0, shN, c2w, Ks, raw, rawcol);
    gs_sort<<<1, 1024, 0, stream>>>(raw, order);
    gs_reorder<<<(NG + 255) / 256, 256, 0, stream>>>(raw, rawcol, order, sp, Bmat);
    // 25600 pixels / 16 per wave = 1600 waves; 8 waves per 256-thread block
    gs_render<<<200, 256, 0, stream>>>(sp, Bmat, out);
}